// TinyTemporalMemoryModel_64707977282154
// MI455X (gfx1250) — compile-verified
//
#include <hip/hip_runtime.h>
#include <cstdint>

// ---------------------------------------------------------------------------
// TinyTemporalMemoryModel on gfx1250:
//   1) bitonic argsort of timestamps (u64 composite keys, L2-resident)
//   2) single-wave sequential scan; per-event GEMVs as 4 chained
//      V_WMMA_F32_16X16X4_F32; operand staging via ds_bpermute with
//      loop-invariant lane addressing; fast native activations;
//      software-pipelined event loads.
// ---------------------------------------------------------------------------

typedef __attribute__((ext_vector_type(2))) float v2f;
typedef __attribute__((ext_vector_type(8))) float v8f;

__device__ __forceinline__ float rdlane_f(float v, int lane) {
  return __int_as_float(__builtin_amdgcn_readlane(__float_as_int(v), lane));
}
__device__ __forceinline__ float bperm_f(int byteaddr, float v) {
  return __int_as_float(__builtin_amdgcn_ds_bpermute(byteaddr, __float_as_int(v)));
}
// fast native activations: v_exp_f32 / v_rcp_f32, no Newton iteration
#define LOG2E 1.4426950408889634f
__device__ __forceinline__ float sig_fast(float x) {
  return __builtin_amdgcn_rcpf(1.0f + __builtin_amdgcn_exp2f(-x * LOG2E));
}
__device__ __forceinline__ float tanh_fast(float x) {
  // tanh(x) = 2*sigmoid(2x) - 1
  return __builtin_fmaf(2.0f, sig_fast(2.0f * x), -1.0f);
}

// --------------------------- sort kernels ----------------------------------

__global__ void init_keys_kernel(const float* __restrict__ ts,
                                 uint64_t* __restrict__ keys, int e, int n) {
  int i = blockIdx.x * blockDim.x + threadIdx.x;
  if (i >= n) return;
  if (i < e) {
    uint32_t u = __float_as_uint(ts[i]);
    u = (u & 0x80000000u) ? ~u : (u | 0x80000000u);  // order-preserving map
    keys[i] = (((uint64_t)u) << 32) | (uint32_t)i;   // tie-break by index
  } else {
    keys[i] = ~0ull;                                 // pad sorts to the end
  }
}

__global__ void bitonic_pass_kernel(uint64_t* __restrict__ keys,
                                    int j, int k, int n) {
  int i = blockIdx.x * blockDim.x + threadIdx.x;
  if (i >= n) return;
  int ixj = i ^ j;
  if (ixj > i) {
    uint64_t a = keys[i], b = keys[ixj];
    bool up = ((i & k) == 0);
    if ((a > b) == up) { keys[i] = b; keys[ixj] = a; }
  }
}

// all sub-stages with j <= 512 run inside one 1024-element LDS tile
__global__ void __launch_bounds__(1024)
bitonic_fused_kernel(uint64_t* __restrict__ keys, int k, int j0) {
  __shared__ uint64_t sh[1024];
  const int tid  = threadIdx.x;
  const int base = blockIdx.x * 1024;
  sh[tid] = keys[base + tid];
  __syncthreads();
  const int i = base + tid;
  const bool up = ((i & k) == 0);
  for (int j = j0; j > 0; j >>= 1) {
    int p = tid ^ j;
    if (p > tid) {
      uint64_t a = sh[tid], b = sh[p];
      if ((a > b) == up) { sh[tid] = b; sh[p] = a; }
    }
    __syncthreads();
  }
  keys[base + tid] = sh[tid];
}

// --------------------------- sequential scan -------------------------------
//
// Packed 16x16 A matrix (loop invariant, 4 K-chunks of 2 VGPRs each):
//   rows  0..7 : W_ih rows 0..7, cols 0..3 += W_hh rows 0..7   (r,z gates)
//   rows  8..11: W_ih rows 8..11                               (xn)
//   rows 12..15: W_hh rows 8..11 in cols 0..3                  (hn)
// B: col 0 = x_src, col 1 = x_dst, staged from the xv register via bpermute.
//
// Distributed registers (one wave):
//   memreg: lane nd*4+j = mem[nd][j] (lanes 0..11), lane 12+nd = lu[nd]
//   xv    : lanes 0..15 = x_src[0..15], lanes 16..31 = x_dst[0..15]

__global__ void __launch_bounds__(32)
tgn_scan_kernel(const uint64_t* __restrict__ keys,
                const int*   __restrict__ src,   const int*   __restrict__ dst,
                const float* __restrict__ ts,    const float* __restrict__ feat,
                const float* __restrict__ W_lin, const float* __restrict__ b_lin,
                const float* __restrict__ W_time,const float* __restrict__ b_time,
                const float* __restrict__ W_ih,  const float* __restrict__ W_hh,
                const float* __restrict__ b_ih,  const float* __restrict__ b_hh,
                float* __restrict__ out, int e) {
  const int lane = threadIdx.x;
  const int n16  = lane & 15;
  const int half = lane >> 4;
  const int l3   = lane & 3;

  // ---- loop-invariant lane predicates / patterns (hoisted by LICM) ----
  const bool grp_dsel = (lane >= 4 && lane < 8) || (lane >= 16 && lane < 20);
  const bool grp_phi  = (n16 >= 10 && n16 <= 13);   // lanes 10..13, 26..29
  const bool grp_f0   = (n16 == 8);
  const bool grp_f1   = (n16 == 9);
  const bool grp_zero = (n16 >= 14);                // lanes 14,15,30,31
  const int  offs4    = l3 * 4;                     // byte offset within node
  const bool is_l3_0 = (l3 == 0), is_l3_1 = (l3 == 1), is_l3_2 = (l3 == 2);

  // B-fragment gather addresses: chunk c, sub-vgpr v -> xv source lane
  int baddr[4][2];
#pragma unroll
  for (int c = 0; c < 4; ++c) {
#pragma unroll
    for (int v = 0; v < 2; ++v) {
      int K = 4 * c + v + 2 * half;
      int srcl = (n16 == 0) ? K : ((n16 == 1) ? (16 + K) : 14);  // lane14 == 0
      baddr[c][v] = srcl * 4;
    }
  }

  // per-lane time-encoding coefficients for the distributed cos
  int pj = grp_phi ? (n16 - 10) : 0;
  const float wtv = W_time[pj];
  const float btv = b_time[pj];

  // wave-uniform small weights replicated into registers
  float wl[20];
  for (int i = 0; i < 20; ++i) wl[i] = W_lin[i];
  const float bl0 = b_lin[0], bl1 = b_lin[1];
  float brz[8];
  for (int i = 0; i < 8; ++i) brz[i] = b_ih[i] + b_hh[i];
  float bxn[4], bhn[4];
  for (int i = 0; i < 4; ++i) { bxn[i] = b_ih[8 + i]; bhn[i] = b_hh[8 + i]; }

  // A fragments: chunk c, sub-vgpr v holds A[row=n16][col=4c+v+2*half]
  v2f Afrag[4];
#pragma unroll
  for (int c = 0; c < 4; ++c) {
#pragma unroll
    for (int v = 0; v < 2; ++v) {
      int row = n16;
      int col = 4 * c + v + 2 * half;
      float val = 0.0f;
      if (row < 8) {
        if (col < 14) val = W_ih[row * 14 + col];
        if (col < 4)  val += W_hh[row * 4 + col];
      } else if (row < 12) {
        if (col < 14) val = W_ih[row * 14 + col];
      } else {
        if (col < 4)  val = W_hh[(row - 4) * 4 + col];  // W_hh rows 8..11
      }
      Afrag[c][v] = val;
    }
  }

  // distributed carry state: mem in lanes 0..11, last_update in lanes 12..14
  float memreg = 0.0f;

  // ---- software pipeline prologue: load event 0 ----
  uint64_t key0 = keys[0];
  int idx_n = __builtin_amdgcn_readfirstlane((int)(uint32_t)(key0 & 0xffffffffu));
  int s_n   = __builtin_amdgcn_readfirstlane(src[idx_n]);
  int d_n   = __builtin_amdgcn_readfirstlane(dst[idx_n]);
  float t_n  = ts[idx_n];
  float f0_n = feat[idx_n * 2 + 0];
  float f1_n = feat[idx_n * 2 + 1];

  for (int i = 0; i < e; ++i) {
    const int   idx = idx_n;
    const int   s   = s_n;
    const int   d   = d_n;
    const float t   = t_n;
    const float f0  = f0_n;
    const float f1  = f1_n;

    // prefetch next event (clamped: no divergent control flow)
    {
      int ip = (i + 1 < e) ? (i + 1) : i;
      uint64_t key = keys[ip];
      idx_n = __builtin_amdgcn_readfirstlane((int)(uint32_t)(key & 0xffffffffu));
      s_n   = __builtin_amdgcn_readfirstlane(src[idx_n]);
      d_n   = __builtin_amdgcn_readfirstlane(dst[idx_n]);
      t_n   = ts[idx_n];
      f0_n  = feat[idx_n * 2 + 0];
      f1_n  = feat[idx_n * 2 + 1];
    }

    // ---- gather node state into message layout (1 bpermute) ----
    int sel_m = grp_dsel ? d : s;          // mem lanes: which node
    int sel_p = half ? d : s;              // phi lanes: whose last_update
    int a_mem = sel_m * 16 + offs4;        // byte addr of mem[sel][l3]
    int a_phi = 48 + sel_p * 4;            // byte addr of lu[sel]
    float g = bperm_f(grp_phi ? a_phi : a_mem, memreg);

    // distributed time encoding: cos(w_t*(t-lu)+b_time) on phi lanes
    float phi = __cosf(__builtin_fmaf(wtv, t - g, btv));

    float xv = g;                 // lanes 0..7 / 16..23: sm|dm / dm|sm
    xv = grp_phi  ? phi  : xv;    // lanes 10..13 / 26..29
    xv = grp_f0   ? f0   : xv;    // lanes 8, 24
    xv = grp_f1   ? f1   : xv;    // lanes 9, 25
    xv = grp_zero ? 0.0f : xv;    // lanes 14,15,30,31 (lane14 = B zero source)

    // ---- 4 chained f32 WMMAs: D[16x16] = A * [x_s | x_d | 0 ...] ----
    v8f acc = {};
#pragma unroll
    for (int c = 0; c < 4; ++c) {
      v2f B;
      B[0] = bperm_f(baddr[c][0], xv);
      B[1] = bperm_f(baddr[c][1], xv);
      acc = __builtin_amdgcn_wmma_f32_16x16x4_f32(
          false, Afrag[c], false, B, (short)0, acc, false, false);
    }

    // ---- pull D columns 0 (src) / 1 (dst) back as wave-uniform scalars ----
    float Ds[16], Dd[16];
#pragma unroll
    for (int r = 0; r < 8; ++r) {
      Ds[r]     = rdlane_f(acc[r], 0);   // rows 0..7,  N=0
      Dd[r]     = rdlane_f(acc[r], 1);   // rows 0..7,  N=1
      Ds[8 + r] = rdlane_f(acc[r], 16);  // rows 8..15, N=0
      Dd[8 + r] = rdlane_f(acc[r], 17);  // rows 8..15, N=1
    }
    float smu[4], dmu[4];
#pragma unroll
    for (int j = 0; j < 4; ++j) {
      smu[j] = rdlane_f(xv, j);          // x_s[0..3] = mem[s]
      dmu[j] = rdlane_f(xv, 4 + j);      // x_s[4..7] = mem[d]
    }

    // ---- GRU gates (wave-uniform math, fast native activations) ----
    float new_sm[4], new_dm[4];
#pragma unroll
    for (int j = 0; j < 4; ++j) {
      float rs = sig_fast(Ds[j] + brz[j]);
      float zs = sig_fast(Ds[4 + j] + brz[4 + j]);
      float ns = tanh_fast(Ds[8 + j] + bxn[j] + rs * (Ds[12 + j] + bhn[j]));
      new_sm[j] = (1.0f - zs) * ns + zs * smu[j];

      float rd = sig_fast(Dd[j] + brz[j]);
      float zd = sig_fast(Dd[4 + j] + brz[4 + j]);
      float nd = tanh_fast(Dd[8 + j] + bxn[j] + rd * (Dd[12 + j] + bhn[j]));
      new_dm[j] = (1.0f - zd) * nd + zd * dmu[j];
    }

    // logit head: W_lin @ [sm, dm, feat] + b_lin (memory BEFORE update)
    float lg0 = bl0 + wl[8] * f0 + wl[9] * f1;
    float lg1 = bl1 + wl[18] * f0 + wl[19] * f1;
#pragma unroll
    for (int j = 0; j < 4; ++j) {
      lg0 += wl[j] * smu[j] + wl[4 + j] * dmu[j];
      lg1 += wl[10 + j] * smu[j] + wl[14 + j] * dmu[j];
    }

    // ---- distributed state update (dst write wins when s == d) ----
    float nsm = is_l3_0 ? new_sm[0] : (is_l3_1 ? new_sm[1]
                                               : (is_l3_2 ? new_sm[2] : new_sm[3]));
    float ndm = is_l3_0 ? new_dm[0] : (is_l3_1 ? new_dm[1]
                                               : (is_l3_2 ? new_dm[2] : new_dm[3]));
    bool w_s  = (lane < 12) && ((lane >> 2) == s);
    bool w_d  = (lane < 12) && ((lane >> 2) == d);
    bool w_lu = (lane == 12 + s) || (lane == 12 + d);
    memreg = w_s  ? nsm : memreg;
    memreg = w_d  ? ndm : memreg;   // after s: d wins, matching .at[s]...at[d]
    memreg = w_lu ? t   : memreg;

    if (lane == 0) {
      out[idx * 2 + 0] = lg0;
      out[idx * 2 + 1] = lg1;
    }
  }
}

// --------------------------- launcher --------------------------------------

extern "C" void kernel_launch(void* const* d_in, const int* in_sizes, int n_in,
                              void* d_out, int out_size, void* d_ws, size_t ws_size,
                              hipStream_t stream) {
  const int*   src    = (const int*)d_in[0];
  const int*   dst    = (const int*)d_in[1];
  const float* ts     = (const float*)d_in[2];
  const float* feat   = (const float*)d_in[3];
  const float* W_lin  = (const float*)d_in[4];
  const float* b_lin  = (const float*)d_in[5];
  const float* W_time = (const float*)d_in[6];
  const float* b_time = (const float*)d_in[7];
  const float* W_ih   = (const float*)d_in[8];
  const float* W_hh   = (const float*)d_in[9];
  const float* b_ih   = (const float*)d_in[10];
  const float* b_hh   = (const float*)d_in[11];

  const int e = in_sizes[0];            // 1,000,000
  int n = 1;
  while (n < e) n <<= 1;                // 2^20 = 1,048,576; 8 MB of keys

  uint64_t* keys = (uint64_t*)d_ws;

  init_keys_kernel<<<(n + 255) / 256, 256, 0, stream>>>(ts, keys, e, n);

  if (n >= 1024) {
    for (int k = 2; k <= n; k <<= 1) {
      int j = k >> 1;
      for (; j > 512; j >>= 1)
        bitonic_pass_kernel<<<(n + 255) / 256, 256, 0, stream>>>(keys, j, k, n);
      bitonic_fused_kernel<<<n / 1024, 1024, 0, stream>>>(keys, k, j);
    }
  } else {
    for (int k = 2; k <= n; k <<= 1)
      for (int j = k >> 1; j > 0; j >>= 1)
        bitonic_pass_kernel<<<(n + 255) / 256, 256, 0, stream>>>(keys, j, k, n);
  }

  tgn_scan_kernel<<<1, 32, 0, stream>>>(keys, src, dst, ts, feat,
                                        W_lin, b_lin, W_time, b_time,
                                        W_ih, W_hh, b_ih, b_hh,
                                        (float*)d_out, e);
}